// MatchLoss_63969242906674
// MI455X (gfx1250) — compile-verified
//
#include <hip/hip_runtime.h>
#include <hip/hip_bf16.h>
#include <float.h>

typedef __attribute__((ext_vector_type(2))) float v2f;
typedef __attribute__((ext_vector_type(8))) float v8f;

#define EPS_F      1e-07f
#define PD_EPS_F   1e-06f
#define RADIUS_F   0.001f

#define BLOCK   128          // 4 waves
#define WPB     4            // waves per block (src tiles per block)
#define CHUNK   1024         // tgt points staged per chunk
#define CTILES  (CHUNK / 16) // 64 tiles per chunk
#define PFPT    (CHUNK / BLOCK) // 8 points staged per thread

// Branchless top-2 insert via flattened selects only (pure v_cmp + v_cndmask,
// no fminf canonicalize noise). Invariant: d1 <= d2.
__device__ __forceinline__ void top2_insert(float& d1, int& i1, float& d2, int& i2,
                                            float d, int i) {
    const bool lt1 = d < d1;
    const bool lt2 = d < d2;
    const float dm = lt1 ? d1 : d;   // value entering slot 2 (if lt2)
    const int   im = lt1 ? i1 : i;   // index entering slot 2 (if lt2)
    d2 = lt2 ? dm : d2;
    i2 = lt2 ? im : i2;
    d1 = lt1 ? d  : d1;
    i1 = lt1 ? i  : i1;
}

// One 16-col tile: build B, WMMA Gram tile, top-2 update. b2n precomputed.
__device__ __forceinline__ void process_tile(
    const v2f& A, float bx, float by, float bz, float b2n, bool hiHalf, int col,
    const float a2v[8], float d1[8], int i1[8], float d2[8], int i2[8]) {

    // B (4x16 f32): lanes 0-15 hold N=col, K=0,1 (x,y); lanes 16-31 K=2,3 (z,0)
    v2f B;
    B[0] = hiHalf ? bz : bx;
    B[1] = hiHalf ? 0.0f : by;

    v8f c = {};
    c = __builtin_amdgcn_wmma_f32_16x16x4_f32(
            false, A, false, B, (short)0, c, false, false);

#pragma unroll
    for (int v = 0; v < 8; ++v) {
        float d = fmaf(-2.0f, c[v], a2v[v] + b2n);   // squared distance
        top2_insert(d1[v], i1[v], d2[v], i2[v], d, col);
    }
}

// One wave owns a 16-row src tile. tgt is swept through double-buffered LDS
// chunks of 1024 float4 (x,y,z,|b|^2); V_WMMA_F32_16X16X4_F32 per 16x16 tile.
__global__ __launch_bounds__(BLOCK) void match_top2_kernel(
    const float* __restrict__ src, const float* __restrict__ tgt,
    float* __restrict__ match_val, int nsrc, int ntgt) {

    __shared__ float4 sbuf[2][CHUNK];   // 32 KB

    const int tid  = threadIdx.x;
    const int lane = tid & 31;
    const int wave = tid >> 5;
    const int srcBase = (blockIdx.x * WPB + wave) * 16;

    const bool hiHalf = (lane >= 16);
    const int col16 = lane & 15;

    // ---- A (16x4 f32): lanes 0-15 row m=lane K=0,1 (x,y); lanes 16-31 K=2,3 (z,0)
    int arow = srcBase + col16;
    int arc = arow < nsrc ? arow : (nsrc - 1);
    float ax = src[3 * arc + 0];
    float ay = src[3 * arc + 1];
    float az = src[3 * arc + 2];
    v2f A;
    A[0] = hiHalf ? az : ax;
    A[1] = hiHalf ? 0.0f : ay;

    // ---- |a|^2 for this lane's 8 accumulator rows
    //      (VGPR v: lanes 0-15 -> M=v, lanes 16-31 -> M=v+8)
    const int rowOff = hiHalf ? 8 : 0;
    float a2v[8];
#pragma unroll
    for (int v = 0; v < 8; ++v) {
        int m = srcBase + rowOff + v;
        int mc = m < nsrc ? m : (nsrc - 1);
        float x = src[3 * mc + 0];
        float y = src[3 * mc + 1];
        float z = src[3 * mc + 2];
        a2v[v] = fmaf(x, x, fmaf(y, y, z * z));
    }

    float d1[8], d2[8];
    int   i1[8], i2[8];
#pragma unroll
    for (int v = 0; v < 8; ++v) {
        d1[v] = FLT_MAX; d2[v] = FLT_MAX;
        i1[v] = 0;       i2[v] = 0;
    }

    const int nfullc = ntgt / CHUNK;    // full staged chunks

    // ---- Stage chunk 0
    if (nfullc > 0) {
#pragma unroll
        for (int k = 0; k < PFPT; ++k) {
            int g = tid + k * BLOCK;
            const float* q = tgt + 3 * g;
            float x = q[0], y = q[1], z = q[2];
            sbuf[0][g] = make_float4(x, y, z, fmaf(x, x, fmaf(y, y, z * z)));
        }
        __syncthreads();
    }

    // ---- Chunk loop: prefetch c+1 into regs, process c from LDS, store c+1.
    for (int c = 0; c < nfullc; ++c) {
        float px[PFPT], py[PFPT], pz[PFPT];
        const bool more = (c + 1) < nfullc;
        if (more) {
            const float* nb = tgt + 3 * ((c + 1) * CHUNK + tid);
#pragma unroll
            for (int k = 0; k < PFPT; ++k) {
                px[k] = nb[3 * k * BLOCK + 0];
                py[k] = nb[3 * k * BLOCK + 1];
                pz[k] = nb[3 * k * BLOCK + 2];
            }
        }

        const float4* buf = sbuf[c & 1];
        const int colBase = c * CHUNK + col16;
#pragma unroll 4
        for (int t = 0; t < CTILES; ++t) {
            float4 q = buf[t * 16 + col16];
            process_tile(A, q.x, q.y, q.z, q.w, hiHalf, colBase + t * 16,
                         a2v, d1, i1, d2, i2);
        }
        __syncthreads();

        if (more) {
            float4* nbuf = sbuf[(c + 1) & 1];
#pragma unroll
            for (int k = 0; k < PFPT; ++k) {
                nbuf[tid + k * BLOCK] =
                    make_float4(px[k], py[k], pz[k],
                                fmaf(px[k], px[k], fmaf(py[k], py[k], pz[k] * pz[k])));
            }
        }
        __syncthreads();
    }

    // ---- Tail targets (ntgt % CHUNK), straight from global, masked.
    const int done = nfullc * CHUNK;
    const int remTiles = (ntgt - done + 15) / 16;
    for (int t = 0; t < remTiles; ++t) {
        int col = done + t * 16 + col16;
        int colc = col < ntgt ? col : (ntgt - 1);
        const float* q = tgt + 3 * colc;
        float bx = q[0], by = q[1], bz = q[2];
        float b2n = fmaf(bx, bx, fmaf(by, by, bz * bz));
        v2f B;
        B[0] = hiHalf ? bz : bx;
        B[1] = hiHalf ? 0.0f : by;
        v8f c = {};
        c = __builtin_amdgcn_wmma_f32_16x16x4_f32(
                false, A, false, B, (short)0, c, false, false);
        const bool colValid = col < ntgt;
#pragma unroll
        for (int v = 0; v < 8; ++v) {
            float d = fmaf(-2.0f, c[v], a2v[v] + b2n);
            d = colValid ? d : FLT_MAX;
            top2_insert(d1[v], i1[v], d2[v], i2[v], d, col);
        }
    }

    // ---- Merge top-2 across the 16 lanes of each half-wave.
#pragma unroll
    for (int off = 8; off >= 1; off >>= 1) {
#pragma unroll
        for (int v = 0; v < 8; ++v) {
            float pd1 = __shfl_xor(d1[v], off);
            int   pi1 = __shfl_xor(i1[v], off);
            float pd2 = __shfl_xor(d2[v], off);
            int   pi2 = __shfl_xor(i2[v], off);
            top2_insert(d1[v], i1[v], d2[v], i2[v], pd1, pi1);
            top2_insert(d1[v], i1[v], d2[v], i2[v], pd2, pi2);
        }
    }

    // ---- Lane 0 writes rows 0..7, lane 16 writes rows 8..15.
    if (col16 == 0) {
#pragma unroll
        for (int v = 0; v < 8; ++v) {
            int m = srcBase + rowOff + v;
            if (m < nsrc) {
                int j = i2[v];                     // second-nearest target
                float dx = src[3 * m + 0] - tgt[3 * j + 0] + PD_EPS_F;
                float dy = src[3 * m + 1] - tgt[3 * j + 1] + PD_EPS_F;
                float dz = src[3 * m + 2] - tgt[3 * j + 2] + PD_EPS_F;
                float pd = sqrtf(fmaf(dx, dx, fmaf(dy, dy, dz * dz)));
                match_val[m] = (pd < RADIUS_F ? 1.0f : 0.0f) + EPS_F;
            }
        }
    }
}

// softplus(logsumexp(match_val)) over all n entries, single block.
__global__ __launch_bounds__(256) void lse_softplus_kernel(
    const float* __restrict__ mv, float* __restrict__ out, int n) {
    __shared__ float sdata[256];
    const int tid = threadIdx.x;

    float m = -FLT_MAX;
    for (int i = tid; i < n; i += 256) m = fmaxf(m, mv[i]);
    sdata[tid] = m;
    __syncthreads();
    for (int s = 128; s > 0; s >>= 1) {
        if (tid < s) sdata[tid] = fmaxf(sdata[tid], sdata[tid + s]);
        __syncthreads();
    }
    const float mx = sdata[0];
    __syncthreads();

    float sum = 0.0f;
    for (int i = tid; i < n; i += 256) sum += expf(mv[i] - mx);
    sdata[tid] = sum;
    __syncthreads();
    for (int s = 128; s > 0; s >>= 1) {
        if (tid < s) sdata[tid] += sdata[tid + s];
        __syncthreads();
    }

    if (tid == 0) {
        float lse = mx + logf(sdata[0]);
        float sp = fmaxf(lse, 0.0f) + log1pf(expf(-fabsf(lse)));  // stable softplus
        out[0] = sp;
    }
}

extern "C" void kernel_launch(void* const* d_in, const int* in_sizes, int n_in,
                              void* d_out, int out_size, void* d_ws, size_t ws_size,
                              hipStream_t stream) {
    const float* src = (const float*)d_in[0];
    const float* tgt = (const float*)d_in[1];
    float* out = (float*)d_out;
    float* mv  = (float*)d_ws;   // nsrc floats of scratch

    const int nsrc = in_sizes[0] / 3;
    const int ntgt = in_sizes[1] / 3;

    const int grid = (nsrc + (16 * WPB) - 1) / (16 * WPB);  // 4 src tiles / block
    match_top2_kernel<<<grid, BLOCK, 0, stream>>>(src, tgt, mv, nsrc, ntgt);
    lse_softplus_kernel<<<1, 256, 0, stream>>>(mv, out, nsrc);
}